// GraphConvolution_35476429865416
// MI455X (gfx1250) — compile-verified
//
#include <hip/hip_runtime.h>

typedef __attribute__((ext_vector_type(2))) float v2f;
typedef __attribute__((ext_vector_type(8))) float v8f;
typedef int v4i_async __attribute__((__vector_size__(4 * sizeof(int))));

#define D_FEAT 128
#define ASTRIDE 132           // padded LDS row stride (floats) -> conflict-free A tile reads
#define WSTRIDE 132           // padded LDS row stride for W^T tile
#define ROWS_PER_BLOCK 64

#if defined(__has_builtin)
#if __has_builtin(__builtin_amdgcn_global_load_async_to_lds_b128)
#define GCN_HAVE_ASYNC_LDS 1
#endif
#endif

#define AS1 __attribute__((address_space(1)))
#define AS3 __attribute__((address_space(3)))

__device__ __forceinline__ unsigned rotl32(unsigned x, int r) {
  return (x << r) | (x >> (32u - r));
}

// Threefry-2x32, 20 rounds, key = jax.random.key(42) -> (0, 42)
__device__ __forceinline__ void threefry2x32_0_42(unsigned x0, unsigned x1,
                                                  unsigned &o0, unsigned &o1) {
  const unsigned ks0 = 0u;
  const unsigned ks1 = 42u;
  const unsigned ks2 = 0x1BD11BDAu ^ 0u ^ 42u;
  x0 += ks0; x1 += ks1;
#define TF_ROUND(r) { x0 += x1; x1 = rotl32(x1, (r)); x1 ^= x0; }
  TF_ROUND(13) TF_ROUND(15) TF_ROUND(26) TF_ROUND(6)
  x0 += ks1; x1 += ks2 + 1u;
  TF_ROUND(17) TF_ROUND(29) TF_ROUND(16) TF_ROUND(24)
  x0 += ks2; x1 += ks0 + 2u;
  TF_ROUND(13) TF_ROUND(15) TF_ROUND(26) TF_ROUND(6)
  x0 += ks0; x1 += ks1 + 3u;
  TF_ROUND(17) TF_ROUND(29) TF_ROUND(16) TF_ROUND(24)
  x0 += ks1; x1 += ks2 + 4u;
  TF_ROUND(13) TF_ROUND(15) TF_ROUND(26) TF_ROUND(6)
  x0 += ks2; x1 += ks0 + 5u;
#undef TF_ROUND
  o0 = x0; o1 = x1;
}

// JAX dropout(p=0.5): bernoulli(key42, 0.5) keep -> x*2, else 0.
__device__ __forceinline__ float dropout_apply(float f, unsigned idx, unsigned H) {
  unsigned j = (idx < H) ? idx : (idx - H);
  unsigned o0, o1;
  threefry2x32_0_42(j, j + H, o0, o1);
  unsigned bits = (idx < H) ? o0 : o1;
  float u = __uint_as_float((bits >> 9) | 0x3f800000u) - 1.0f;  // [0,1)
  return (u < 0.5f) ? (f * 2.0f) : 0.0f;
}

// Guaranteed hardware f32 atomic add at device scope (non-returning -> STOREcnt;
// S_ENDPGM's implicit wait-idle covers completion).
__device__ __forceinline__ void atomicAddF32(float* p, float v) {
  asm volatile("global_atomic_add_f32 %0, %1, off scope:SCOPE_DEV"
               :
               : "v"(p), "v"(v)
               : "memory");
}

// -------- Kernel 0: Wt[n][k] = W[k][n] so B tiles load as single b64 --------
__global__ __launch_bounds__(256) void gcn_transpose_w(
    const float* __restrict__ W, float* __restrict__ WT) {
  int idx = blockIdx.x * 256 + threadIdx.x;   // grid covers 128*128
  int k = idx >> 7;
  int n = idx & 127;
  WT[n * D_FEAT + k] = W[idx];
}

// -------- Kernel 1: support = dropout(feats) @ W via V_WMMA_F32_16X16X4_F32 --
// Block = 256 threads = 8 waves; block covers 64 rows. Wave w: row-tile (w>>1),
// column half (w&1) -> 16x64 output stripe = 4 accumulators of v8f.
// W^T is async-copied global->LDS (ASYNCcnt path), hidden under the threefry
// dropout burst; B tiles then come from the LDS pipe, conflict-free.
__global__ __launch_bounds__(256) void gcn_gemm_dropout(
    const float* __restrict__ feats, const float* __restrict__ WT,
    float* __restrict__ support, int N, unsigned H) {
  __shared__ float sA[ROWS_PER_BLOCK * ASTRIDE];
#if defined(GCN_HAVE_ASYNC_LDS)
  __shared__ float sWT[D_FEAT * WSTRIDE];
#endif
  const int t = threadIdx.x;
  const int rowBase = blockIdx.x * ROWS_PER_BLOCK;

#if defined(GCN_HAVE_ASYNC_LDS)
  // Issue async W^T copy first: 4096 16B chunks, padded rows in LDS.
#pragma unroll
  for (int i = 0; i < 16; ++i) {
    int flat = i * 256 + t;            // 16B chunk 0..4095
    int n = flat >> 5;                 // W^T row 0..127
    int kc = (flat & 31) << 2;         // k column 0..124
    __builtin_amdgcn_global_load_async_to_lds_b128(
        (AS1 v4i_async*)(WT + n * D_FEAT + kc),
        (AS3 v4i_async*)(sWT + n * WSTRIDE + kc), 0, 0);
  }
#endif

  // Stage 64x128 A tile into LDS with dropout fused (coalesced float4 loads).
#pragma unroll
  for (int i = 0; i < 8; ++i) {
    int fidx = i * 256 + t;          // float4 index 0..2047
    int lr = fidx >> 5;              // local row 0..63
    int c4 = (fidx & 31) << 2;       // column 0..124
    int grow = rowBase + lr;
    int crow = (grow < N) ? grow : (N - 1);  // clamp; OOB rows never stored
    const float4 v = *(const float4*)(feats + (size_t)crow * D_FEAT + c4);
    unsigned base = (unsigned)crow * D_FEAT + (unsigned)c4;
    float* dst = sA + lr * ASTRIDE + c4;
    dst[0] = dropout_apply(v.x, base + 0u, H);
    dst[1] = dropout_apply(v.y, base + 1u, H);
    dst[2] = dropout_apply(v.z, base + 2u, H);
    dst[3] = dropout_apply(v.w, base + 3u, H);
  }
#if defined(GCN_HAVE_ASYNC_LDS)
  asm volatile("s_wait_asynccnt 0x0" ::: "memory");
#endif
  __syncthreads();

  const int wave = t >> 5;
  const int lane = t & 31;
  const int r  = wave >> 1;   // row tile 0..3
  const int ch = wave & 1;    // column half 0..1
  const int m  = lane & 15;
  const int h  = lane >> 4;

  v8f acc[4] = {};
  // A 16x4 fp32 layout: lane half h, VGPR v -> K = 4*kb + 2*h + v
  const float* aPtr = sA + (16 * r + m) * ASTRIDE + 2 * h;
  const int n0 = 64 * ch + m;

#pragma unroll
  for (int kb = 0; kb < 32; ++kb) {
    const int k = 4 * kb + 2 * h;
    v2f a = *(const v2f*)(aPtr + 4 * kb);
#pragma unroll
    for (int nt = 0; nt < 4; ++nt) {
      const int n = n0 + 16 * nt;
      // B 4x16 fp32 layout: lane half h, VGPR v -> K = v + 2*h.
      // W^T[n][k], W^T[n][k+1] adjacent -> single 8-byte load.
#if defined(GCN_HAVE_ASYNC_LDS)
      v2f b = *(const v2f*)(sWT + n * WSTRIDE + k);
#else
      v2f b = *(const v2f*)(WT + n * D_FEAT + k);
#endif
      acc[nt] = __builtin_amdgcn_wmma_f32_16x16x4_f32(
          false, a, false, b, (short)0, acc[nt], false, false);
    }
  }

  // C/D 16x16 fp32 layout: VGPR v, lane half h -> M = v + 8*h, N = lane&15
#pragma unroll
  for (int nt = 0; nt < 4; ++nt) {
    const int col = 64 * ch + 16 * nt + m;
#pragma unroll
    for (int v = 0; v < 8; ++v) {
      const int grow = rowBase + 16 * r + v + 8 * h;
      if (grow < N) support[(size_t)grow * D_FEAT + col] = acc[nt][v];
    }
  }
}

// -------- Kernel 2: out[n][:] = bias (rows with no edges must equal bias) ----
__global__ __launch_bounds__(256) void gcn_init_bias(
    float4* __restrict__ out4, const float4* __restrict__ bias4, int total4) {
  int i = blockIdx.x * blockDim.x + threadIdx.x;
  if (i < total4) out4[i] = bias4[i & 31];
}

// -------- Kernel 3: SpMM scatter-add. One wave per edge. ---------------------
// Edge metadata scalarized via readfirstlane -> SMEM loads; coalesced 512B
// float4 gather of support[col]; hardware global_atomic_add_f32 x4 per lane;
// software prefetch of next iteration's gather row (random-address, L2-resident).
__global__ __launch_bounds__(256) void gcn_spmm_atomic(
    const int* __restrict__ row_idx, const int* __restrict__ col_idx,
    const float* __restrict__ val, const float* __restrict__ support,
    float* __restrict__ out, int E) {
  const int lane = threadIdx.x & 31;
  int w = (int)((blockIdx.x * blockDim.x + threadIdx.x) >> 5);
  const int nw = (int)((gridDim.x * blockDim.x) >> 5);
  for (int e = w; e < E; e += nw) {
    const int eu = __builtin_amdgcn_readfirstlane(e);   // wave-uniform -> s_load
    const int r = row_idx[eu];
    const int c = col_idx[eu];
    const float vv = val[eu];

    // Prefetch next edge's gather row while we work on this one.
    const int en = eu + nw;
    if (en < E) {
      const int cn = __builtin_amdgcn_readfirstlane(col_idx[en]);
      __builtin_prefetch(support + (size_t)cn * D_FEAT + lane * 4, 0, 0);
    }

    const float4 s = *(const float4*)(support + (size_t)c * D_FEAT + lane * 4);
    float* o = out + (size_t)r * D_FEAT + lane * 4;
    atomicAddF32(o + 0, vv * s.x);
    atomicAddF32(o + 1, vv * s.y);
    atomicAddF32(o + 2, vv * s.z);
    atomicAddF32(o + 3, vv * s.w);
  }
}

extern "C" void kernel_launch(void* const* d_in, const int* in_sizes, int n_in,
                              void* d_out, int out_size, void* d_ws, size_t ws_size,
                              hipStream_t stream) {
  const float* feats   = (const float*)d_in[0];
  const int*   adj_row = (const int*)d_in[1];
  const int*   adj_col = (const int*)d_in[2];
  const float* adj_val = (const float*)d_in[3];
  const float* weight  = (const float*)d_in[4];
  const float* bias    = (const float*)d_in[5];
  float* out = (float*)d_out;

  const int N = in_sizes[0] / D_FEAT;
  const int E = in_sizes[1];
  const unsigned H = (unsigned)(in_sizes[0] / 2);  // JAX iota half-split point

  float* support = (float*)d_ws;                       // N*128 f32 (51.2 MB)
  float* WT      = support + (size_t)N * D_FEAT;       // 128*128 f32 (64 KB)

  gcn_transpose_w<<<(D_FEAT * D_FEAT) / 256, 256, 0, stream>>>(weight, WT);

  const int gemmBlocks = (N + ROWS_PER_BLOCK - 1) / ROWS_PER_BLOCK;
  gcn_gemm_dropout<<<gemmBlocks, 256, 0, stream>>>(feats, WT, support, N, H);

  const int total4 = N * (D_FEAT / 4);
  gcn_init_bias<<<(total4 + 255) / 256, 256, 0, stream>>>(
      (float4*)out, (const float4*)bias, total4);

  gcn_spmm_atomic<<<2048, 256, 0, stream>>>(adj_row, adj_col, adj_val,
                                            support, out, E);
}